// GIN2_47940424958476
// MI455X (gfx1250) — compile-verified
//
#include <hip/hip_runtime.h>
#include <hip/hip_bf16.h>

#define NNODES 100000
#define NEDGES 1600000
#define FEAT 64
#define NLAYERS 3
#define NGRAPH 1000
#define BN_EPS 1e-5f

typedef __attribute__((ext_vector_type(2))) float v2f;
typedef __attribute__((ext_vector_type(8))) float v8f;

__device__ __forceinline__ float atomAddGlobalF(float* p, float v) {
  return __hip_atomic_fetch_add(p, v, __ATOMIC_RELAXED, __HIP_MEMORY_SCOPE_AGENT);
}
__device__ __forceinline__ float atomAddLdsF(float* p, float v) {
  return __hip_atomic_fetch_add(p, v, __ATOMIC_RELAXED, __HIP_MEMORY_SCOPE_WORKGROUP);
}
__device__ __forceinline__ void atomMaxF(float* addr, float val) {
  unsigned int* ua = (unsigned int*)addr;
  unsigned int old = __hip_atomic_load(ua, __ATOMIC_RELAXED, __HIP_MEMORY_SCOPE_AGENT);
  while (__uint_as_float(old) < val) {
    unsigned int assumed = old;
    old = atomicCAS(ua, assumed, __float_as_uint(val));
    if (old == assumed) break;
  }
}

__global__ void fill_f32(float* __restrict__ p, long n, float v) {
  long i = (long)blockIdx.x * blockDim.x + threadIdx.x;
  if (i < n) p[i] = v;
}

// agg[dst[e]][f] += h[src[e]][f]; float4 per thread: b128 gathers, 4x fewer index reads.
// Node table (25.6MB) is L2-resident (192MB L2) -> this is an L2-bandwidth/atomic phase.
__global__ void scatter_add_k(const float* __restrict__ h, int lda,
                              const int* __restrict__ src,
                              const int* __restrict__ dst,
                              float* __restrict__ agg, long total) {
  long t = (long)blockIdx.x * blockDim.x + threadIdx.x;   // total = E*16
  if (t >= total) return;
  int e  = (int)(t >> 4);
  int f0 = (int)(t & 15) * 4;
  int s = src[e], d = dst[e];
  const float4 hv = *(const float4*)(h + (size_t)s * lda + f0);
  float* ap = agg + (size_t)d * FEAT + f0;
  atomAddGlobalF(ap + 0, hv.x);
  atomAddGlobalF(ap + 1, hv.y);
  atomAddGlobalF(ap + 2, hv.z);
  atomAddGlobalF(ap + 3, hv.w);
}

// One wave -> 16x64 output tile via V_WMMA_F32_16X16X4_F32 (K steps of 4).
// A frag (16x4 f32): lanes0-15 K={k,k+1}, lanes16-31 K={k+2,k+3}  -> one b64 load.
// B frags kept fragment-major in LDS: sB[(kstep*32+lane)*8 + nt*2 + v] so each lane
// reads its 8 B floats per k-step as two contiguous 16B LDS loads (no repack movs).
// Compile-time flags kill the per-iteration EXEC masking seen in round 0.
template <bool HAS_AGG, bool STATS, bool RELU>
__launch_bounds__(256)
__global__ void gemm_wmma_k(const float* __restrict__ A, int lda,
                            const float* __restrict__ Agg,
                            const float* __restrict__ epsp, int layer,
                            const float* __restrict__ W, int K,
                            const float* __restrict__ bias,
                            float* __restrict__ Y, int ldy,
                            float* __restrict__ colsum,
                            float* __restrict__ colsq,
                            int mtiles) {
  __shared__ float sB[192 * 64];     // K<=192: fragment-major W
  __shared__ float sSum[64];
  __shared__ float sSq[64];
  const int tid = threadIdx.x;
  // Fragment-major fill: i -> (kstep, lane, nt, v)
  for (int i = tid; i < K * 64; i += 256) {
    int s_  = i >> 8;           // kstep
    int r   = i & 255;
    int ln  = r >> 3;
    int j   = r & 7;
    int nt  = j >> 1;
    int v   = j & 1;
    int k   = 4 * s_ + 2 * (ln >> 4) + v;
    int n   = nt * 16 + (ln & 15);
    sB[i] = W[k * 64 + n];
  }
  if (STATS && tid < 64) { sSum[tid] = 0.0f; sSq[tid] = 0.0f; }
  __syncthreads();

  const int wave = tid >> 5;
  const int lane = tid & 31;
  const int tile = blockIdx.x * 8 + wave;   // wave-uniform guard: EXEC all-ones for WMMA
  if (tile < mtiles) {
    const int hi = lane >> 4;
    const int nl = lane & 15;
    const int row = tile * 16 + nl;
    const float* arow = A + (size_t)row * lda + 2 * hi;
    const float* grow = Agg + (size_t)row * FEAT + 2 * hi;
    const float hscale = HAS_AGG ? (1.0f + epsp[layer]) : 1.0f;

    v8f acc0 = {}, acc1 = {}, acc2 = {}, acc3 = {};
    const int ksteps = K >> 2;
    for (int ks = 0; ks < ksteps; ++ks) {
      v2f a = *(const v2f*)(arow + 4 * ks);
      if (HAS_AGG) {
        v2f g = *(const v2f*)(grow + 4 * ks);
        a.x = __builtin_fmaf(hscale, a.x, g.x);
        a.y = __builtin_fmaf(hscale, a.y, g.y);
      }
      const float* bb = &sB[(ks * 32 + lane) * 8];
      v2f b0 = *(const v2f*)(bb + 0);
      v2f b1 = *(const v2f*)(bb + 2);
      v2f b2 = *(const v2f*)(bb + 4);
      v2f b3 = *(const v2f*)(bb + 6);
      acc0 = __builtin_amdgcn_wmma_f32_16x16x4_f32(false, a, false, b0, (short)0, acc0, false, false);
      acc1 = __builtin_amdgcn_wmma_f32_16x16x4_f32(false, a, false, b1, (short)0, acc1, false, false);
      acc2 = __builtin_amdgcn_wmma_f32_16x16x4_f32(false, a, false, b2, (short)0, acc2, false, false);
      acc3 = __builtin_amdgcn_wmma_f32_16x16x4_f32(false, a, false, b3, (short)0, acc3, false, false);
    }

    v8f accs[4] = {acc0, acc1, acc2, acc3};
    #pragma unroll
    for (int nt = 0; nt < 4; ++nt) {
      const int n0 = nt * 16;
      const float bv = bias[n0 + nl];
      union { v8f v; float f[8]; } u;
      u.v = accs[nt];
      float ps = 0.0f, pq = 0.0f;
      #pragma unroll
      for (int i = 0; i < 8; ++i) {       // C/D: VGPR i -> row i + 8*hi, col n0+nl
        float y = u.f[i] + bv;
        if (RELU) y = fmaxf(y, 0.0f);
        Y[(size_t)(tile * 16 + hi * 8 + i) * ldy + n0 + nl] = y;
        ps += y; pq += y * y;
      }
      if (STATS) {
        atomAddLdsF(&sSum[n0 + nl], ps);
        atomAddLdsF(&sSq[n0 + nl], pq);
      }
    }
  }
  __syncthreads();
  if (STATS && tid < 64) {
    atomAddGlobalF(&colsum[tid], sSum[tid]);
    atomAddGlobalF(&colsq[tid], sSq[tid]);
  }
}

__global__ void bn_finalize_k(const float* __restrict__ cs, const float* __restrict__ cq,
                              const float* __restrict__ g, const float* __restrict__ be,
                              float invN, float* __restrict__ scale, float* __restrict__ shift) {
  int f = threadIdx.x;
  if (f < 64) {
    float mu  = cs[f] * invN;
    float var = cq[f] * invN - mu * mu;     // biased variance
    float s = g[f] * rsqrtf(var + BN_EPS);
    scale[f] = s;
    shift[f] = be[f] - mu * s;
  }
}

// z = relu(y*scale + shift), float4 per thread, strided output (ldo = 64 or 192)
__global__ void bn_apply_relu_k(const float* __restrict__ y,
                                const float* __restrict__ scale,
                                const float* __restrict__ shift,
                                float* __restrict__ out, int ldo, long total) {
  long t = (long)blockIdx.x * blockDim.x + threadIdx.x;   // total = N*16
  if (t >= total) return;
  int n  = (int)(t >> 4);
  int f0 = (int)(t & 15) * 4;
  float4 yv = *(const float4*)(y + 4 * t);
  float4 sc = *(const float4*)(scale + f0);
  float4 sh = *(const float4*)(shift + f0);
  float4 o;
  o.x = fmaxf(fmaf(yv.x, sc.x, sh.x), 0.0f);
  o.y = fmaxf(fmaf(yv.y, sc.y, sh.y), 0.0f);
  o.z = fmaxf(fmaf(yv.z, sc.z, sh.z), 0.0f);
  o.w = fmaxf(fmaf(yv.w, sc.w, sh.w), 0.0f);
  *(float4*)(out + (size_t)n * ldo + f0) = o;
}

__global__ void gate_k(const float* __restrict__ hh, const float* __restrict__ gW,
                       const float* __restrict__ gb, float* __restrict__ gate, int n) {
  int i = blockIdx.x * blockDim.x + threadIdx.x;
  if (i >= n) return;
  const float* r = hh + (size_t)i * 64;
  float acc = 0.0f;
  #pragma unroll
  for (int f = 0; f < 64; ++f) acc = fmaf(r[f], gW[f], acc);
  gate[i] = acc + gb[0];
}

__global__ void segmax_k(const float* __restrict__ gate, const int* __restrict__ bidx,
                         float* __restrict__ m, int n) {
  int i = blockIdx.x * blockDim.x + threadIdx.x;
  if (i >= n) return;
  atomMaxF(&m[bidx[i]], gate[i]);
}

__global__ void segexp_k(const float* __restrict__ gate, const int* __restrict__ bidx,
                         const float* __restrict__ m, float* __restrict__ ebuf,
                         float* __restrict__ denom, int n) {
  int i = blockIdx.x * blockDim.x + threadIdx.x;
  if (i >= n) return;
  int b = bidx[i];
  float e = expf(gate[i] - m[b]);
  ebuf[i] = e;
  atomAddGlobalF(&denom[b], e);
}

// pooled[b] += hh[n]*alpha[n], float4 per thread
__global__ void pool_k(const float* __restrict__ hh, const float* __restrict__ ebuf,
                       const float* __restrict__ denom, const int* __restrict__ bidx,
                       float* __restrict__ pooled, long total) {
  long t = (long)blockIdx.x * blockDim.x + threadIdx.x;   // total = N*16
  if (t >= total) return;
  int n  = (int)(t >> 4);
  int f0 = (int)(t & 15) * 4;
  int b = bidx[n];
  float alpha = ebuf[n] / denom[b];
  float4 hv = *(const float4*)(hh + 4 * t);
  float* pp = pooled + (size_t)b * 64 + f0;
  atomAddGlobalF(pp + 0, hv.x * alpha);
  atomAddGlobalF(pp + 1, hv.y * alpha);
  atomAddGlobalF(pp + 2, hv.z * alpha);
  atomAddGlobalF(pp + 3, hv.w * alpha);
}

extern "C" void kernel_launch(void* const* d_in, const int* in_sizes, int n_in,
                              void* d_out, int out_size, void* d_ws, size_t ws_size,
                              hipStream_t stream) {
  (void)in_sizes; (void)n_in; (void)out_size; (void)ws_size;
  const float* x    = (const float*)d_in[0];
  const int*   eidx = (const int*)d_in[1];
  const int*   src  = eidx;
  const int*   dst  = eidx + NEDGES;
  const int*   bidx = (const int*)d_in[2];
  const float* eps  = (const float*)d_in[3];
  const float* W1   = (const float*)d_in[4];
  const float* b1   = (const float*)d_in[5];
  const float* g1   = (const float*)d_in[6];
  const float* be1  = (const float*)d_in[7];
  const float* W2   = (const float*)d_in[8];
  const float* b2   = (const float*)d_in[9];
  const float* g2   = (const float*)d_in[10];
  const float* be2  = (const float*)d_in[11];
  const float* l1W  = (const float*)d_in[12];
  const float* l1b  = (const float*)d_in[13];
  const float* gW   = (const float*)d_in[14];
  const float* gb   = (const float*)d_in[15];

  float* out = (float*)d_out;
  float* hh = out;                                 // [N,64]
  float* pooled = out + (size_t)NNODES * 64;       // [NG,64]

  float* ws = (float*)d_ws;
  size_t o = 0;
  float* agg  = ws + o; o += (size_t)NNODES * 64;
  float* tmp  = ws + o; o += (size_t)NNODES * 64;
  float* zb   = ws + o; o += (size_t)NNODES * 64;
  float* xs   = ws + o; o += (size_t)NNODES * 192; // concat buffer [N, 3*64]
  float* gate = ws + o; o += NNODES;
  float* ebuf = ws + o; o += NNODES;
  float* mbuf = ws + o; o += NGRAPH;
  float* den  = ws + o; o += NGRAPH;
  float* csum = ws + o; o += 64;
  float* csq  = ws + o; o += 64;                   // contiguous with csum
  float* bsc  = ws + o; o += 64;
  float* bsh  = ws + o; o += 64;

  const int TPB = 256;
  const int mtiles = NNODES / 16;              // 6250 exact
  const int gemm_blocks = (mtiles + 7) / 8;    // 8 waves / block
  const long nf  = (long)NNODES * 64;
  const long nv4 = (long)NNODES * 16;          // float4 granularity
  const long ev4 = (long)NEDGES * 16;

  for (int l = 0; l < NLAYERS; ++l) {
    const float* hptr = (l == 0) ? x : (xs + 64 * (l - 1));
    const int hld = (l == 0) ? 64 : 192;

    fill_f32<<<(int)((nf + TPB - 1) / TPB), TPB, 0, stream>>>(agg, nf, 0.0f);
    scatter_add_k<<<(int)((ev4 + TPB - 1) / TPB), TPB, 0, stream>>>(hptr, hld, src, dst, agg, ev4);

    fill_f32<<<1, 128, 0, stream>>>(csum, 128, 0.0f);
    gemm_wmma_k<true, true, false><<<gemm_blocks, TPB, 0, stream>>>(hptr, hld, agg, eps, l,
        W1 + (size_t)l * 64 * 64, 64, b1 + l * 64, tmp, 64, csum, csq, mtiles);
    bn_finalize_k<<<1, 64, 0, stream>>>(csum, csq, g1 + l * 64, be1 + l * 64,
        1.0f / (float)NNODES, bsc, bsh);
    bn_apply_relu_k<<<(int)((nv4 + TPB - 1) / TPB), TPB, 0, stream>>>(tmp, bsc, bsh, zb, 64, nv4);

    fill_f32<<<1, 128, 0, stream>>>(csum, 128, 0.0f);
    gemm_wmma_k<false, true, false><<<gemm_blocks, TPB, 0, stream>>>(zb, 64, nullptr, nullptr, 0,
        W2 + (size_t)l * 64 * 64, 64, b2 + l * 64, tmp, 64, csum, csq, mtiles);
    bn_finalize_k<<<1, 64, 0, stream>>>(csum, csq, g2 + l * 64, be2 + l * 64,
        1.0f / (float)NNODES, bsc, bsh);
    bn_apply_relu_k<<<(int)((nv4 + TPB - 1) / TPB), TPB, 0, stream>>>(tmp, bsc, bsh, xs + 64 * l, 192, nv4);
  }

  // hh = relu(hcat @ lin1_W + lin1_b), K=192, written straight to d_out
  gemm_wmma_k<false, false, true><<<gemm_blocks, TPB, 0, stream>>>(xs, 192, nullptr, nullptr, 0,
      l1W, 192, l1b, hh, 64, nullptr, nullptr, mtiles);

  gate_k<<<(NNODES + TPB - 1) / TPB, TPB, 0, stream>>>(hh, gW, gb, gate, NNODES);
  fill_f32<<<(NGRAPH + TPB - 1) / TPB, TPB, 0, stream>>>(mbuf, NGRAPH, -3.0e38f);
  fill_f32<<<(NGRAPH + TPB - 1) / TPB, TPB, 0, stream>>>(den, NGRAPH, 0.0f);
  fill_f32<<<(int)(((long)NGRAPH * 64 + TPB - 1) / TPB), TPB, 0, stream>>>(pooled, (long)NGRAPH * 64, 0.0f);
  segmax_k<<<(NNODES + TPB - 1) / TPB, TPB, 0, stream>>>(gate, bidx, mbuf, NNODES);
  segexp_k<<<(NNODES + TPB - 1) / TPB, TPB, 0, stream>>>(gate, bidx, mbuf, ebuf, den, NNODES);
  pool_k<<<(int)((nv4 + TPB - 1) / TPB), TPB, 0, stream>>>(hh, ebuf, den, bidx, pooled, nv4);
}